// SeedGAT_3504693313907
// MI455X (gfx1250) — compile-verified
//
#include <hip/hip_runtime.h>
#include <hip/hip_bf16.h>

// ---------------------------------------------------------------------------
// SeedGAT on MI455X (gfx1250): WMMA f32 GEMMs + atomic segment softmax/scatter
// ---------------------------------------------------------------------------

#define N_NODES 50000
#define N_EDGES 1600000
#define HC      128          // HEADS*HID
#define HEADS   4
#define HID     32
#define SEEDD   128
#define NEG_SLOPE 0.2f

typedef float v2f __attribute__((ext_vector_type(2)));
typedef float v8f __attribute__((ext_vector_type(8)));

__device__ __forceinline__ float lrelu(float x) {
    return x > 0.0f ? x : NEG_SLOPE * x;
}

// float atomic max via signed/unsigned bit-pattern trick (valid for all IEEE
// floats given the location is initialized with a real value, which it is:
// the self-loop logit).
__device__ __forceinline__ void atomicMaxF(float* addr, float val) {
    if (val >= 0.0f) {
        atomicMax((int*)addr, __float_as_int(val));
    } else {
        atomicMin((unsigned int*)addr, __float_as_uint(val));
    }
}

// ---------------------------------------------------------------------------
// Generic fp32 WMMA GEMM:  C[M,Ncols] = act(A[M,K] * B[K,Ncols] + bias)
// grid.x = M/16 (M multiple of 16), blockDim.x = 32 * (Ncols/16), K mult of 4,
// K <= 256. Uses V_WMMA_F32_16X16X4_F32 per K-step of 4.
// ---------------------------------------------------------------------------
__global__ void __launch_bounds__(256)
gemm_wmma_f32(const float* __restrict__ A, const float* __restrict__ B,
              const float* __restrict__ bias, float* __restrict__ C,
              int K, int Ncols, int relu_act) {
    __shared__ float sA[16 * 256];

    const int tid = threadIdx.x;
    const int nthreads = blockDim.x;

    // A strip for 16 consecutive rows is contiguous: copy with float4
    const float4* Ag = (const float4*)(A + (long)blockIdx.x * 16 * K);
    float4* Sv = (float4*)sA;
    const int nvec = (16 * K) >> 2;
    for (int i = tid; i < nvec; i += nthreads) Sv[i] = Ag[i];
    __syncthreads();

    const int wave = tid >> 5;
    const int lane = tid & 31;
    const int half = lane >> 4;   // selects K-pair within a 4-wide K step
    const int m    = lane & 15;   // M index for A frag, N index for B/C frags
    const int colbase = wave << 4;

    v8f acc = {};
    for (int kb = 0; kb < K; kb += 4) {
        const int k0 = kb + 2 * half;
        v2f a, b;
        // A 16x4 (MxK): VGPR0 -> K = k0, VGPR1 -> K = k0+1, lane = M
        a.x = sA[m * K + k0];
        a.y = sA[m * K + k0 + 1];
        // B 4x16 (KxN): row striped across lanes within a VGPR, halves = K+2
        b.x = B[(long)k0 * Ncols + colbase + m];
        b.y = B[(long)(k0 + 1) * Ncols + colbase + m];
        acc = __builtin_amdgcn_wmma_f32_16x16x4_f32(
            /*neg_a=*/false, a, /*neg_b=*/false, b,
            /*c_mod=*/(short)0, acc, /*reuse_a=*/false, /*reuse_b=*/false);
    }

    const float bv = bias ? bias[colbase + m] : 0.0f;
    const long row0 = (long)blockIdx.x * 16;
#pragma unroll
    for (int r = 0; r < 8; ++r) {
        // C/D 16x16 f32: VGPR r -> M = r (lanes 0-15) / r+8 (lanes 16-31)
        const int row = r + 8 * half;
        float v = acc[r] + bv;
        if (relu_act) v = fmaxf(v, 0.0f);
        C[(row0 + row) * (long)Ncols + colbase + m] = v;
    }
}

// ---------------------------------------------------------------------------
// Per-(node,head): a_src/a_dst dot products + self-loop logit as max-init
// ---------------------------------------------------------------------------
__global__ void node_prep(const float* __restrict__ h,
                          const float* __restrict__ att_src,
                          const float* __restrict__ att_dst,
                          float* __restrict__ asrc, float* __restrict__ adst,
                          float* __restrict__ mbuf, int n) {
    const int t = blockIdx.x * blockDim.x + threadIdx.x;
    if (t >= n * HEADS) return;
    const int node = t >> 2, hd = t & 3;
    const float* hp = h + (long)node * HC + hd * HID;
    const float* as = att_src + hd * HID;
    const float* ad = att_dst + hd * HID;
    float s = 0.0f, d = 0.0f;
#pragma unroll
    for (int c = 0; c < HID; ++c) {
        const float v = hp[c];
        s += v * as[c];
        d += v * ad[c];
    }
    asrc[t] = s;
    adst[t] = d;
    mbuf[t] = lrelu(s + d);   // self-loop edge logit seeds the segment max
}

// ---------------------------------------------------------------------------
// Edge pass 1: atomic segment max of leaky-relu logits (4 heads per thread)
// ---------------------------------------------------------------------------
__global__ void edge_max(const int* __restrict__ src, const int* __restrict__ dst,
                         const float* __restrict__ asrc,
                         const float* __restrict__ adst,
                         float* __restrict__ mbuf, int ne) {
    const int e = blockIdx.x * blockDim.x + threadIdx.x;
    if (e >= ne) return;
    const int s = src[e], d = dst[e];
    const float4 a = *(const float4*)(asrc + 4 * s);
    const float4 b = *(const float4*)(adst + 4 * d);
    atomicMaxF(&mbuf[4 * d + 0], lrelu(a.x + b.x));
    atomicMaxF(&mbuf[4 * d + 1], lrelu(a.y + b.y));
    atomicMaxF(&mbuf[4 * d + 2], lrelu(a.z + b.z));
    atomicMaxF(&mbuf[4 * d + 3], lrelu(a.w + b.w));
}

// ---------------------------------------------------------------------------
// Denominator init with the self-loop term: den = exp(e_self - m)
// ---------------------------------------------------------------------------
__global__ void denom_init(const float* __restrict__ asrc,
                           const float* __restrict__ adst,
                           const float* __restrict__ mbuf,
                           float* __restrict__ den, int n) {
    const int t = blockIdx.x * blockDim.x + threadIdx.x;
    if (t >= n * HEADS) return;
    const float e = lrelu(asrc[t] + adst[t]);
    den[t] = __expf(e - mbuf[t]);
}

// ---------------------------------------------------------------------------
// Edge pass 2: atomic segment sum of exp(e - m[dst])
// ---------------------------------------------------------------------------
__global__ void edge_denom(const int* __restrict__ src, const int* __restrict__ dst,
                           const float* __restrict__ asrc,
                           const float* __restrict__ adst,
                           const float* __restrict__ mbuf,
                           float* __restrict__ den, int ne) {
    const int e = blockIdx.x * blockDim.x + threadIdx.x;
    if (e >= ne) return;
    const int s = src[e], d = dst[e];
    const float4 a = *(const float4*)(asrc + 4 * s);
    const float4 b = *(const float4*)(adst + 4 * d);
    const float4 m = *(const float4*)(mbuf + 4 * d);
    atomicAdd(&den[4 * d + 0], __expf(lrelu(a.x + b.x) - m.x));
    atomicAdd(&den[4 * d + 1], __expf(lrelu(a.y + b.y) - m.y));
    atomicAdd(&den[4 * d + 2], __expf(lrelu(a.z + b.z) - m.z));
    atomicAdd(&den[4 * d + 3], __expf(lrelu(a.w + b.w) - m.w));
}

// ---------------------------------------------------------------------------
// Out init with the self-loop message: out = h * alpha_self
// Thread handles one float4 (4 cols, single head). n*32 threads.
// ---------------------------------------------------------------------------
__global__ void out_init(const float* __restrict__ h,
                         const float* __restrict__ asrc,
                         const float* __restrict__ adst,
                         const float* __restrict__ mbuf,
                         const float* __restrict__ den,
                         float* __restrict__ out, int n) {
    const int t = blockIdx.x * blockDim.x + threadIdx.x;
    if (t >= n * (HC / 4)) return;
    const int node = t >> 5, q = t & 31;
    const int hd = q >> 3;               // 8 float4's per head (32 cols)
    const int idx = node * HEADS + hd;
    const float e = lrelu(asrc[idx] + adst[idx]);
    const float alpha = __expf(e - mbuf[idx]) / (den[idx] + 1e-16f);
    const float4 hv = *(const float4*)(h + (long)node * HC + q * 4);
    float4 o;
    o.x = hv.x * alpha; o.y = hv.y * alpha; o.z = hv.z * alpha; o.w = hv.w * alpha;
    *(float4*)(out + (long)node * HC + q * 4) = o;
}

// ---------------------------------------------------------------------------
// Edge pass 3 (the heavy one): wave32 per edge, lane -> 4 columns.
// read h[src] (512B coalesced), 128 f32 atomic adds into out[dst] (L2).
// ---------------------------------------------------------------------------
__global__ void __launch_bounds__(256)
edge_scatter(const int* __restrict__ src, const int* __restrict__ dst,
             const float* __restrict__ h,
             const float* __restrict__ asrc, const float* __restrict__ adst,
             const float* __restrict__ mbuf, const float* __restrict__ den,
             float* __restrict__ out, int ne) {
    const int gw = (blockIdx.x * blockDim.x + threadIdx.x) >> 5;
    if (gw >= ne) return;
    const int lane = threadIdx.x & 31;
    const int s = src[gw], d = dst[gw];
    const int hd = lane >> 3;
    const int di = d * HEADS + hd;
    const float e = lrelu(asrc[s * HEADS + hd] + adst[di]);
    const float alpha = __expf(e - mbuf[di]) / (den[di] + 1e-16f);
    const float4 hv = *(const float4*)(h + (long)s * HC + lane * 4);
    float* op = out + (long)d * HC + lane * 4;
    atomicAdd(op + 0, hv.x * alpha);
    atomicAdd(op + 1, hv.y * alpha);
    atomicAdd(op + 2, hv.z * alpha);
    atomicAdd(op + 3, hv.w * alpha);
}

// ---------------------------------------------------------------------------
// cur = (out_p + bias_p) - (out_n + bias_n) + (residual ? cur : 0)
// ---------------------------------------------------------------------------
__global__ void combine(const float* __restrict__ outp, const float* __restrict__ outn,
                        const float* __restrict__ bp, const float* __restrict__ bn,
                        float* __restrict__ cur, int n, int residual) {
    const int t = blockIdx.x * blockDim.x + threadIdx.x;
    if (t >= n * (HC / 4)) return;
    const long i = (long)t * 4;
    const int c = (int)(i & (HC - 1));
    const float4 p = *(const float4*)(outp + i);
    const float4 q = *(const float4*)(outn + i);
    const float4 bpv = *(const float4*)(bp + c);
    const float4 bnv = *(const float4*)(bn + c);
    float4 r;
    if (residual) r = *(const float4*)(cur + i);
    else { r.x = 0.f; r.y = 0.f; r.z = 0.f; r.w = 0.f; }
    r.x += (p.x + bpv.x) - (q.x + bnv.x);
    r.y += (p.y + bpv.y) - (q.y + bnv.y);
    r.z += (p.z + bpv.z) - (q.z + bnv.z);
    r.w += (p.w + bpv.w) - (q.w + bnv.w);
    *(float4*)(cur + i) = r;
}

// ---------------------------------------------------------------------------
// feats = concat(cur[N,128], seed[N,128]) -> [N,256]
// ---------------------------------------------------------------------------
__global__ void concat_feats(const float* __restrict__ cur,
                             const float* __restrict__ seed,
                             float* __restrict__ feats, int n) {
    const int t = blockIdx.x * blockDim.x + threadIdx.x;
    if (t >= n * 64) return;           // 64 float4 per output row
    const int node = t >> 6, q = t & 63;
    float4 v;
    if (q < 32) v = *(const float4*)(cur  + (long)node * HC + q * 4);
    else        v = *(const float4*)(seed + (long)node * SEEDD + (q - 32) * 4);
    *(float4*)(feats + (long)node * (HC + SEEDD) + q * 4) = v;
}

// ---------------------------------------------------------------------------
// Host-side launcher
// ---------------------------------------------------------------------------
extern "C" void kernel_launch(void* const* d_in, const int* in_sizes, int n_in,
                              void* d_out, int out_size, void* d_ws, size_t ws_size,
                              hipStream_t stream) {
    (void)in_sizes; (void)n_in; (void)out_size; (void)ws_size;
    const int n = N_NODES, ne = N_EDGES;

    const float* x      = (const float*)d_in[0];
    const int*   pos_ei = (const int*)d_in[1];     // [2,E]: src | dst
    const int*   neg_ei = (const int*)d_in[2];
    const float* seed   = (const float*)d_in[3];

    // params flattened in dict-insertion order:
    // pos[0..2]{W, att_src, att_dst, bias}, neg[0..2]{...}, pred{W1,b1,W2,b2}
    struct Layer { const float *W, *as, *ad, *b; };
    Layer pos[3], neg[3];
    int p = 4;
    for (int i = 0; i < 3; ++i) {
        pos[i].W  = (const float*)d_in[p++]; pos[i].as = (const float*)d_in[p++];
        pos[i].ad = (const float*)d_in[p++]; pos[i].b  = (const float*)d_in[p++];
    }
    for (int i = 0; i < 3; ++i) {
        neg[i].W  = (const float*)d_in[p++]; neg[i].as = (const float*)d_in[p++];
        neg[i].ad = (const float*)d_in[p++]; neg[i].b  = (const float*)d_in[p++];
    }
    const float* W1 = (const float*)d_in[p++];
    const float* b1 = (const float*)d_in[p++];
    const float* W2 = (const float*)d_in[p++];
    const float* b2 = (const float*)d_in[p++];

    // workspace partition (floats)
    float* ws   = (float*)d_ws;
    float* cur  = ws;                  // [N,128] running node features
    float* hbuf = cur  + (long)n * HC; // [N,128] per-conv transformed feats
    float* outp = hbuf + (long)n * HC; // [N,128]
    float* outn = outp + (long)n * HC; // [N,128]
    float* asrc = outn + (long)n * HC; // [N,4]
    float* adst = asrc + (long)n * 4;  // [N,4]
    float* mbuf = adst + (long)n * 4;  // [N,4]
    float* den  = mbuf + (long)n * 4;  // [N,4]
    float* feats= den  + (long)n * 4;  // [N,256]
    float* hmid = feats+ (long)n * (HC + SEEDD); // [N,32]

    const int B = 256;
    const int gN4  = (n * 4  + B - 1) / B;
    const int gN32 = (n * 32 + B - 1) / B;
    const int gE   = (ne + B - 1) / B;
    const int gE32 = (ne * 32 + B - 1) / B;  // wave-per-edge scatter

    auto conv = [&](const float* input, const Layer& L,
                    const int* srcI, const int* dstI, float* outbuf) {
        gemm_wmma_f32<<<n / 16, 32 * (HC / 16), 0, stream>>>(
            input, L.W, nullptr, hbuf, 128, HC, 0);
        node_prep<<<gN4, B, 0, stream>>>(hbuf, L.as, L.ad, asrc, adst, mbuf, n);
        edge_max<<<gE, B, 0, stream>>>(srcI, dstI, asrc, adst, mbuf, ne);
        denom_init<<<gN4, B, 0, stream>>>(asrc, adst, mbuf, den, n);
        edge_denom<<<gE, B, 0, stream>>>(srcI, dstI, asrc, adst, mbuf, den, ne);
        out_init<<<gN32, B, 0, stream>>>(hbuf, asrc, adst, mbuf, den, outbuf, n);
        edge_scatter<<<gE32, B, 0, stream>>>(srcI, dstI, hbuf, asrc, adst, mbuf,
                                             den, outbuf, ne);
    };

    const float* inp = x;
    for (int i = 0; i < 3; ++i) {
        conv(inp, pos[i], pos_ei, pos_ei + ne, outp);
        conv(inp, neg[i], neg_ei, neg_ei + ne, outn);
        combine<<<gN32, B, 0, stream>>>(outp, outn, pos[i].b, neg[i].b, cur, n,
                                        i > 0 ? 1 : 0);
        inp = cur;
    }

    // prediction head
    concat_feats<<<(n * 64 + B - 1) / B, B, 0, stream>>>(cur, seed, feats, n);
    gemm_wmma_f32<<<n / 16, 32 * (HID / 16), 0, stream>>>(
        feats, W1, b1, hmid, HC + SEEDD, HID, 1);
    gemm_wmma_f32<<<n / 16, 32 * (16 / 16), 0, stream>>>(
        hmid, W2, b2, (float*)d_out, HID, 16, 0);
}